// DiarizationModel_41326175322751
// MI455X (gfx1250) — compile-verified
//
#include <hip/hip_runtime.h>
#include <math.h>

typedef float v2f __attribute__((ext_vector_type(2)));
typedef float v8f __attribute__((ext_vector_type(8)));

#define LAMBDA1  0.3366f
#define LR       0.1f
#define SHRINK_C (LR * LAMBDA1)
#define ADAM_B1  0.9f
#define ADAM_B2  0.999f
#define ADAM_EPS 1e-8f
#define N_ITERS  10

__device__ __forceinline__ float sgnf(float x) {
  return (float)((x > 0.f) - (x < 0.f));
}
__device__ __forceinline__ float shrinkf(float x) {
  // faithful to the reference's buggy shrink: sign(x) * max(0, |x - c|)
  return sgnf(x) * fmaxf(0.f, fabsf(x - SHRINK_C));
}

// ---------------- init: P = basis, m = v = 0 ----------------
__global__ void k_init(float* __restrict__ P, float* __restrict__ M,
                       float* __restrict__ V, const float* __restrict__ basis,
                       int n) {
  int i = blockIdx.x * blockDim.x + threadIdx.x;
  if (i < n) { P[i] = basis[i]; M[i] = 0.f; V[i] = 0.f; }
}

// ---------------- prep: B = first ? P : shrink(P); reset argmax ----------------
__global__ void k_prep(float* __restrict__ B, const float* __restrict__ P, int n,
                       int first, unsigned long long* __restrict__ packed) {
  int i = blockIdx.x * blockDim.x + threadIdx.x;
  if (i < n) B[i] = first ? P[i] : shrinkf(P[i]);
  if (blockIdx.x == 0 && threadIdx.x == 0) *packed = 0ull;
}

// ---------------- fused GEMM + column-abs-sum + argmax over T columns ----------
// One wave per 16-column tile. C tile (16x16 f32) built from 4x
// V_WMMA_F32_16X16X4_F32 per 16-row block of the 512-row output; |E - C|
// accumulated per column directly in the C-fragment layout.
__global__ __launch_bounds__(256) void k_colmax(const float* __restrict__ E,
                                                const float* __restrict__ A,
                                                const float* __restrict__ B,
                                                unsigned long long* __restrict__ packed,
                                                int T) {
  constexpr int D = 512;
  __shared__ float bt[16 * D];  // B transposed: bt[k*D + i] = B[i][k]  (32 KB)
  const int tid = threadIdx.x;
  for (int idx = tid; idx < D * 16; idx += 256)
    bt[(idx & 15) * D + (idx >> 4)] = B[idx];
  __syncthreads();

  const int lane = tid & 31;
  const int half = lane >> 4;   // lanes 16..31 handle the "other" K / row half
  const int l    = lane & 15;   // column-in-tile / row-in-block index
  const int gwave  = blockIdx.x * 8 + (tid >> 5);
  const int nwaves = gridDim.x * 8;
  const int ntiles = T >> 4;

  for (int tile = gwave; tile < ntiles; tile += nwaves) {
    const int j = (tile << 4) + l;

    // B-operand fragments: activation rows (K) x 16 cols, K-slice s covers 4s..4s+3
    // reg0: lanes0-15 -> K=4s, lanes16-31 -> K=4s+1 ; reg1: K=4s+2 / 4s+3
    v2f bf[4];
#pragma unroll
    for (int s = 0; s < 4; ++s) {
      const int k0 = 4 * s + half;
      bf[s].x = A[(size_t)k0 * T + j];
      bf[s].y = A[(size_t)(k0 + 2) * T + j];
    }
    if (tile + nwaves < ntiles)
      __builtin_prefetch(A + ((size_t)(tile + nwaves) << 4), 0, 1);

    float acc = 0.f;  // partial column abs-sum (this lane's half of the rows)
    for (int i0 = 0; i0 < D; i0 += 16) {
      v8f c = {};
#pragma unroll
      for (int s = 0; s < 4; ++s) {
        // A-operand: basis rows i0..i0+15, K-slice 4s..4s+3
        // lanes0-15: K=4s,4s+1 ; lanes16-31: K=4s+2,4s+3
        const int kk = 4 * s + 2 * half;
        v2f af;
        af.x = bt[kk * D + i0 + l];
        af.y = bt[(kk + 1) * D + i0 + l];
        c = __builtin_amdgcn_wmma_f32_16x16x4_f32(false, af, false, bf[s],
                                                  (short)0, c, false, false);
      }
      // C layout: lane l holds column j; VGPR r = row i0+r (lanes 0-15),
      // row i0+8+r (lanes 16-31). E is row-major (D x T): coalesced per row.
      const float* e = E + (size_t)(i0 + 8 * half) * T + j;
#pragma unroll
      for (int r = 0; r < 8; ++r)
        acc += fabsf(e[(size_t)r * T] - c[r]);
    }

    // full column sum = this half + mirrored half
    float colsum = acc + __shfl_xor(acc, 16, 32);

    // wave-local max + argmax (first-occurrence tie-break)
    float mv = colsum;
    int   mj = j;
#pragma unroll
    for (int off = 8; off >= 1; off >>= 1) {
      float ov = __shfl_xor(mv, off, 32);
      int   oj = __shfl_xor(mj, off, 32);
      if (ov > mv || (ov == mv && oj < mj)) { mv = ov; mj = oj; }
    }
    if (lane == 0) {
      // colsum >= 0 so float bits are monotone; ~j makes smaller j win ties.
      unsigned long long p =
          ((unsigned long long)__float_as_uint(mv) << 32) |
          (unsigned long long)(0xFFFFFFFFu - (unsigned)mj);
      atomicMax(packed, p);
    }
  }
}

// ---------------- gradient + Adam update (single block, O(8K) work) ------------
__global__ __launch_bounds__(256) void k_update(const float* __restrict__ E,
                                                const float* __restrict__ A,
                                                const float* __restrict__ B,
                                                float* __restrict__ P,
                                                float* __restrict__ M,
                                                float* __restrict__ V,
                                                const unsigned long long* __restrict__ packed,
                                                int T, int chain, float bc1, float bc2) {
  constexpr int D = 512, K = 16;
  __shared__ float acol[K];
  __shared__ float srow[D];
  __shared__ float part[256];
  __shared__ int   k2s;
  const int tid = threadIdx.x;

  const int j1 = (int)(0xFFFFFFFFu - (unsigned)(*packed & 0xFFFFFFFFull));
  if (tid < K) acol[tid] = A[(size_t)tid * T + j1];
  __syncthreads();

  // sign of residual column j1 (recompute the single column)
  for (int i = tid; i < D; i += 256) {
    float c = 0.f;
#pragma unroll
    for (int k = 0; k < K; ++k) c += B[i * K + k] * acol[k];
    srow[i] = sgnf(E[(size_t)i * T + j1] - c);
  }
  // column abs-sums of B -> k2
  {
    const int k = tid & 15;
    float p = 0.f;
    for (int i = tid >> 4; i < D; i += 16) p += fabsf(B[i * K + k]);
    part[tid] = p;
  }
  __syncthreads();
  if (tid == 0) {
    float best = -1.f; int bk = 0;
    for (int k = 0; k < K; ++k) {
      float s = 0.f;
      for (int g = 0; g < 16; ++g) s += part[g * 16 + k];
      if (s > best) { best = s; bk = k; }
    }
    k2s = bk;
  }
  __syncthreads();
  const int k2 = k2s;

  for (int idx = tid; idx < D * K; idx += 256) {
    const int i = idx >> 4, k = idx & 15;
    float g = -srow[i] * acol[k];
    if (k == k2) g += LAMBDA1 * sgnf(B[idx]);
    float p = P[idx];
    if (chain) g *= sgnf(p) * sgnf(p - SHRINK_C);  // d shrink / dp (a.e.)
    float m = ADAM_B1 * M[idx] + (1.f - ADAM_B1) * g;
    float v = ADAM_B2 * V[idx] + (1.f - ADAM_B2) * g * g;
    M[idx] = m; V[idx] = v;
    P[idx] = p - LR * (m / bc1) / (sqrtf(v / bc2) + ADAM_EPS);
  }
}

// ---------------- finalize: out = [shrink(P), A] ----------------
__global__ void k_finalize(float* __restrict__ out, const float* __restrict__ P,
                           const float* __restrict__ A, int nB, int nA) {
  int i = blockIdx.x * blockDim.x + threadIdx.x;
  if (i < nB) out[i] = shrinkf(P[i]);
  for (int idx = i; idx < nA; idx += gridDim.x * blockDim.x)
    out[nB + idx] = A[idx];
}

extern "C" void kernel_launch(void* const* d_in, const int* in_sizes, int n_in,
                              void* d_out, int out_size, void* d_ws, size_t ws_size,
                              hipStream_t stream) {
  (void)n_in; (void)out_size; (void)ws_size;
  const float* E     = (const float*)d_in[0];
  const float* basis = (const float*)d_in[1];
  const float* A     = (const float*)d_in[2];
  const int DK = in_sizes[1];       // 512*16
  const int KT = in_sizes[2];       // 16*50000
  const int T  = KT / 16;
  float* out = (float*)d_out;

  unsigned long long* packed = (unsigned long long*)d_ws;
  float* wsf = (float*)d_ws;
  float* P = wsf + 4;               // keep 16B alignment after the u64 slot
  float* M = P + DK;
  float* V = M + DK;
  float* B = V + DK;

  k_init<<<(DK + 255) / 256, 256, 0, stream>>>(P, M, V, basis, DK);

  const int ntiles = T / 16;
  const int nblk   = (ntiles + 7) / 8;   // one wave (of 8/block) per tile
  double p1 = 1.0, p2 = 1.0;
  for (int t = 1; t <= N_ITERS; ++t) {
    p1 *= (double)ADAM_B1; p2 *= (double)ADAM_B2;
    k_prep<<<(DK + 255) / 256, 256, 0, stream>>>(B, P, DK, (t == 1) ? 1 : 0, packed);
    k_colmax<<<nblk, 256, 0, stream>>>(E, A, B, packed, T);
    k_update<<<1, 256, 0, stream>>>(E, A, B, P, M, V, packed, T, (t > 1) ? 1 : 0,
                                    (float)(1.0 - p1), (float)(1.0 - p2));
  }
  k_finalize<<<(KT + 255) / 256, 256, 0, stream>>>(out, P, A, DK, KT);
}